// SIGNConv_39994735460363
// MI455X (gfx1250) — compile-verified
//
#include <hip/hip_runtime.h>
#include <hip/hip_bf16.h>

typedef __attribute__((ext_vector_type(2))) float v2f;
typedef __attribute__((ext_vector_type(8))) float v8f;

#define D_FEAT 128
#define K_DIM  256   // 2 * D_FEAT
#define D_OUT  128

// ---------------------------------------------------------------------------
// Kernel 1: zero agg[N*128] and deg[N] (laid out contiguously in ws)
// ---------------------------------------------------------------------------
__global__ void zero_ws_kernel(float* __restrict__ p, int total) {
    int i = blockIdx.x * blockDim.x + threadIdx.x;
    if (i < total) p[i] = 0.0f;
}

// ---------------------------------------------------------------------------
// Kernel 2: Wt[k*128 + n] = W[n*256 + k]   (W is [D_OUT=128, 2*D_FEAT=256])
// ---------------------------------------------------------------------------
__global__ void transpose_w_kernel(const float* __restrict__ W, float* __restrict__ Wt) {
    int idx = blockIdx.x * blockDim.x + threadIdx.x;   // over 128*256
    if (idx >= D_OUT * K_DIM) return;
    int n = idx >> 8;      // 0..127
    int k = idx & 255;     // 0..255
    Wt[k * D_OUT + n] = W[idx];
}

// ---------------------------------------------------------------------------
// Kernel 3: edge scatter. One wave32 per edge: lane loads float4 of the
// source row (coalesced 512B), atomically adds into agg[dst]. Both arrays
// are L2-resident (25.6MB each vs 192MB L2), so atomics run at L2 rate.
// ---------------------------------------------------------------------------
__global__ __launch_bounds__(256) void edge_scatter_kernel(
    const float* __restrict__ feat, const int* __restrict__ src,
    const int* __restrict__ dst, float* __restrict__ agg,
    float* __restrict__ deg, int n_edges) {
    int gthread = blockIdx.x * blockDim.x + threadIdx.x;
    int e    = gthread >> 5;          // wave index == edge index
    int lane = gthread & 31;
    if (e >= n_edges) return;
    int s = src[e];
    int d = dst[e];
    const float4* frow = (const float4*)(feat + (size_t)s * D_FEAT);
    float4 v = frow[lane];            // 16B per lane, 512B per wave, contiguous
    float* arow = agg + (size_t)d * D_FEAT + lane * 4;
    atomicAdd(arow + 0, v.x);
    atomicAdd(arow + 1, v.y);
    atomicAdd(arow + 2, v.z);
    atomicAdd(arow + 3, v.w);
    if (lane == 0) atomicAdd(deg + d, 1.0f);
}

// ---------------------------------------------------------------------------
// Kernel 4: per-block 16-row M tile. 8 waves, wave w owns N-tile w (16 cols).
// X = [agg/deg | feature] staged in LDS (16x256 = 16KB). GEMM via
// V_WMMA_F32_16X16X4_F32 accumulating K=256 in steps of 4. Then bias,
// cross-wave row sum-of-squares via shfl_xor + ds_add_f32, L2-normalize, store.
//
// Fragment layouts (ISA 7.12.2, f32):
//   A 16x4 : lane l -> row (l&15), K = kb + (l>>4)*2 + vgpr
//   B 4x16 : lane l -> col (l&15), K = kb + (l>>4)*2 + vgpr
//   C 16x16: lane l, vgpr v -> row v + (l>>4)*8, col (l&15)
// ---------------------------------------------------------------------------
__global__ __launch_bounds__(256) void gemm_norm_kernel(
    const float* __restrict__ feat, const float* __restrict__ agg,
    const float* __restrict__ deg,  const float* __restrict__ Wt,
    const float* __restrict__ bias, float* __restrict__ out, int n_nodes) {
    __shared__ float Xs[16 * K_DIM];   // 16 KB
    __shared__ float dinv[16];
    __shared__ float rowsum[16];

    int tid   = threadIdx.x;
    int mbase = blockIdx.x * 16;

    if (tid < 16) {
        int node = mbase + tid;
        float dg = (node < n_nodes) ? deg[node] : 0.0f;
        dinv[tid]   = (dg > 0.0f) ? (1.0f / fmaxf(dg, 1.0f)) : 0.0f;
        rowsum[tid] = 0.0f;
    }
    __syncthreads();

    // Stage 16x256 activation strip: [agg*dinv | feature]
    for (int idx = tid; idx < 16 * K_DIM; idx += 256) {
        int r = idx >> 8;        // row 0..15
        int c = idx & 255;       // col 0..255
        int node = mbase + r;
        float v = 0.0f;
        if (node < n_nodes) {
            if (c < D_FEAT)
                v = agg[(size_t)node * D_FEAT + c] * dinv[r];
            else
                v = feat[(size_t)node * D_FEAT + (c - D_FEAT)];
        }
        Xs[idx] = v;
    }
    __syncthreads();

    int wave = tid >> 5;              // N-tile 0..7
    int lane = tid & 31;
    int l15  = lane & 15;
    int koff = (lane >> 4) * 2;       // 0 for lanes 0-15, 2 for lanes 16-31
    int ncol = wave * 16 + l15;       // output column 0..127

    const float* xrow = Xs + l15 * K_DIM + koff;   // A source (LDS)
    const float* bcol = Wt + ncol;                 // B source (global, L2-hot)

    v8f acc = {};
    #pragma unroll 4
    for (int kb = 0; kb < K_DIM; kb += 4) {
        v2f a, b;
        a.x = xrow[kb];                            // ds_load_b64
        a.y = xrow[kb + 1];
        b.x = bcol[(size_t)(kb + koff) * D_OUT];   // half-wave contiguous
        b.y = bcol[(size_t)(kb + koff + 1) * D_OUT];
        acc = __builtin_amdgcn_wmma_f32_16x16x4_f32(
            /*neg_a=*/false, a, /*neg_b=*/false, b,
            /*c_mod=*/(short)0, acc, /*reuse_a=*/false, /*reuse_b=*/false);
    }

    // Bias: C column depends only on (lane&15) -> one scalar per lane
    float bv = bias[ncol];
    #pragma unroll
    for (int i = 0; i < 8; i++) acc[i] += bv;

    // Row sum-of-squares. Row of acc[i] is i + (lane>>4)*8; its 16 elements in
    // this wave live in one 16-lane half-group -> xor butterfly over 1,2,4,8.
    #pragma unroll
    for (int i = 0; i < 8; i++) {
        float sq = acc[i] * acc[i];
        sq += __shfl_xor(sq, 1, 32);
        sq += __shfl_xor(sq, 2, 32);
        sq += __shfl_xor(sq, 4, 32);
        sq += __shfl_xor(sq, 8, 32);
        if (l15 == 0)   // lane 0 (rows 0-7) and lane 16 (rows 8-15)
            atomicAdd(&rowsum[i + (lane >> 4) * 8], sq);   // ds_add_f32
    }
    __syncthreads();

    // Normalize and store: out = acc / max(||row||, 1e-12)
    int mrow0 = (lane >> 4) * 8;
    #pragma unroll
    for (int i = 0; i < 8; i++) {
        int m    = mrow0 + i;
        int node = mbase + m;
        if (node < n_nodes) {
            float scale = 1.0f / fmaxf(sqrtf(rowsum[m]), 1e-12f);
            out[(size_t)node * D_OUT + ncol] = acc[i] * scale;
        }
    }
}

// ---------------------------------------------------------------------------
extern "C" void kernel_launch(void* const* d_in, const int* in_sizes, int n_in,
                              void* d_out, int out_size, void* d_ws, size_t ws_size,
                              hipStream_t stream) {
    const float* feature = (const float*)d_in[0];
    const int*   src     = (const int*)d_in[1];
    const int*   dst     = (const int*)d_in[2];
    const float* W       = (const float*)d_in[3];
    const float* bias    = (const float*)d_in[4];
    float*       out     = (float*)d_out;

    const int n_nodes = in_sizes[0] / D_FEAT;   // 50000
    const int n_edges = in_sizes[1];            // 1600000

    // Workspace layout: agg[N*128] | deg[N] | Wt[256*128]
    float* agg = (float*)d_ws;
    float* deg = agg + (size_t)n_nodes * D_FEAT;
    float* Wt  = deg + n_nodes;

    // 1) zero agg+deg (contiguous)
    {
        int total = n_nodes * (D_FEAT + 1);
        zero_ws_kernel<<<(total + 255) / 256, 256, 0, stream>>>(agg, total);
    }
    // 2) transpose W -> Wt
    transpose_w_kernel<<<(D_OUT * K_DIM + 255) / 256, 256, 0, stream>>>(W, Wt);
    // 3) edge scatter: one wave32 per edge
    {
        long long threads = (long long)n_edges * 32;
        int blocks = (int)((threads + 255) / 256);
        edge_scatter_kernel<<<blocks, 256, 0, stream>>>(feature, src, dst, agg, deg, n_edges);
    }
    // 4) fused mean + concat-GEMM (f32 WMMA) + bias + L2 normalize
    {
        int blocks = (n_nodes + 15) / 16;       // 3125 M-tiles
        gemm_norm_kernel<<<blocks, 256, 0, stream>>>(feature, agg, deg, Wt, bias, out, n_nodes);
    }
}